// AttentionInteraction_80461917323827
// MI455X (gfx1250) — compile-verified
//
#include <hip/hip_runtime.h>

typedef __attribute__((ext_vector_type(2))) float v2f;
typedef __attribute__((ext_vector_type(8))) float v8f;

#define F_SPARSE 26
#define BATCH    4096
#define EMB_DIM  128
#define ATT_DIM  64
#define NPAIRS   325   // 26*25/2
#define MPAD     32    // feature rows padded to 2 WMMA M-tiles

// One block per batch row. 256 threads = 8 wave32.
// Phase 1: P[0:32, 0:128] = Epad[0:32, 0:128] @ Wcat[0:128, 0:128]  (fp32 WMMA)
//   where Wcat[:, 0:64]   = W1[0:128, :]   (Pa)
//         Wcat[:, 64:128] = W1[128:256, :] (Pb)
// Phase 2: out[b,p] = sum_a relu(Pa[i_p,a] + Pb[j_p,a] + b1[a]) * W2[a] + b2
__global__ __launch_bounds__(256) void attn_interaction_kernel(
    const float* __restrict__ emb,   // [26, 4096, 128]
    const float* __restrict__ W1,    // [256, 64]
    const float* __restrict__ b1,    // [64]
    const float* __restrict__ W2,    // [64, 1]
    const float* __restrict__ b2,    // [1]
    float* __restrict__ out)         // [4096, 325]
{
    __shared__ float Elds[MPAD * EMB_DIM];  // 16 KB
    __shared__ float Plds[MPAD * EMB_DIM];  // 16 KB: cols 0..63 = Pa, 64..127 = Pb

    const int b    = blockIdx.x;
    const int tid  = threadIdx.x;
    const int lane = tid & 31;
    const int wave = tid >> 5;      // 0..7

    // ---- Stage E[b] into LDS: features 0..25 real, 26..31 zero-padded ----
    // Embeddings are streamed once across the whole grid -> non-temporal loads.
    for (int idx = tid; idx < MPAD * EMB_DIM; idx += 256) {
        const int f = idx >> 7;      // /EMB_DIM
        const int d = idx & 127;
        float v = 0.0f;
        if (f < F_SPARSE)
            v = __builtin_nontemporal_load(
                    &emb[(size_t)f * (BATCH * EMB_DIM) + (size_t)b * EMB_DIM + d]);
        Elds[idx] = v;
    }
    __syncthreads();

    // ---- Phase 1: fp32 WMMA GEMM ----
    // wave owns N-tile `wave` (cols wave*16 .. wave*16+15) and both M-tiles.
    {
        const int nl    = (lane & 15);            // N within tile
        const int n     = wave * 16 + nl;         // global N (0..127)
        const int khalf = (lane >> 4) * 2;        // 0 or 2 (K split across lane halves)
        const int m0    = nl;                     // A: M = lane%16 for both halves
        const int m1    = 16 + nl;

        // Branch-free, loop-invariant base for Wcat column n:
        //   Wcat[kk][n] = Wbase[kk * ATT_DIM]
        const float* __restrict__ Wbase =
            W1 + ((n < ATT_DIM) ? n : (EMB_DIM * ATT_DIM + (n - ATT_DIM)));

        const float* __restrict__ Arow0 = &Elds[m0 * EMB_DIM + khalf];
        const float* __restrict__ Arow1 = &Elds[m1 * EMB_DIM + khalf];

        v8f c0 = {};  // M-tile 0 (features 0..15)
        v8f c1 = {};  // M-tile 1 (features 16..31)

        #pragma unroll
        for (int k = 0; k < EMB_DIM; k += 4) {
            // A fragments (16x4 fp32): lane M = lane%16; VGPR0=K=kb, VGPR1=K=kb+1
            // kb = k + khalf is even -> 8-byte-aligned v2f loads (ds b64).
            const v2f a0 = *(const v2f*)(Arow0 + k);
            const v2f a1 = *(const v2f*)(Arow1 + k);
            // B fragment (4x16 fp32): lane N = lane%16; VGPR0=K=kb, VGPR1=K=kb+1
            v2f bf;
            bf.x = Wbase[(k + khalf) * ATT_DIM];
            bf.y = Wbase[(k + khalf + 1) * ATT_DIM];
            // D = A*B + C, fp32 in / fp32 out
            c0 = __builtin_amdgcn_wmma_f32_16x16x4_f32(false, a0, false, bf,
                                                       (short)0, c0, false, false);
            c1 = __builtin_amdgcn_wmma_f32_16x16x4_f32(false, a1, false, bf,
                                                       (short)0, c1, false, false);
        }

        // C/D layout: VGPR v -> M = v + 8*(lane/16), N = lane%16 (within tile)
        const int mrow = (lane >> 4) * 8;
        const int ncol = wave * 16 + nl;
        #pragma unroll
        for (int v = 0; v < 8; ++v) {
            Plds[(mrow + v) * EMB_DIM + ncol]      = c0[v];
            Plds[(16 + mrow + v) * EMB_DIM + ncol] = c1[v];
        }
    }
    __syncthreads();

    // ---- Phase 2: pair interactions, one pair per wave iteration ----
    const float b1v0 = b1[lane];
    const float b1v1 = b1[lane + 32];
    const float w2v0 = W2[lane];
    const float w2v1 = W2[lane + 32];
    const float b2v  = b2[0];

    int p = 0;
    for (int i = 0; i < F_SPARSE - 1; ++i) {
        for (int j = i + 1; j < F_SPARSE; ++j, ++p) {
            if ((p & 7) != wave) continue;   // uniform per wave
            float ha = Plds[i * EMB_DIM + lane]
                     + Plds[j * EMB_DIM + ATT_DIM + lane] + b1v0;
            float hb = Plds[i * EMB_DIM + lane + 32]
                     + Plds[j * EMB_DIM + ATT_DIM + lane + 32] + b1v1;
            ha = fmaxf(ha, 0.0f);
            hb = fmaxf(hb, 0.0f);
            float s = ha * w2v0 + hb * w2v1;
            // wave32 butterfly reduction
            #pragma unroll
            for (int off = 16; off > 0; off >>= 1)
                s += __shfl_xor(s, off, 32);
            if (lane == 0)
                out[(size_t)b * NPAIRS + p] = s + b2v;
        }
    }
}

extern "C" void kernel_launch(void* const* d_in, const int* in_sizes, int n_in,
                              void* d_out, int out_size, void* d_ws, size_t ws_size,
                              hipStream_t stream) {
    const float* emb = (const float*)d_in[0];  // [26, 4096, 128]
    const float* W1  = (const float*)d_in[1];  // [256, 64]
    const float* b1  = (const float*)d_in[2];  // [64]
    const float* W2  = (const float*)d_in[3];  // [64, 1]
    const float* b2  = (const float*)d_in[4];  // [1]
    float* out = (float*)d_out;                // [4096, 325]

    attn_interaction_kernel<<<BATCH, 256, 0, stream>>>(emb, W1, b1, W2, b2, out);
}